// DCNN_15195594293678
// MI455X (gfx1250) — compile-verified
//
#include <hip/hip_runtime.h>
#include <math.h>

// ---------------------------------------------------------------------------
// Deformable-conv CNN forward for MI455X (gfx1250, wave32).
// GEMM-shaped stages use V_WMMA_F32_16X16X4_F32 (full fp32: workload is
// latency/bandwidth bound, ~6 GFLOPs total, so no reason to drop precision).
// ---------------------------------------------------------------------------

typedef __attribute__((ext_vector_type(2))) float v2f;
typedef __attribute__((ext_vector_type(8))) float v8f;

#define NB   512
#define H1   28
#define W1   28
#define C1O  32
#define H2   14
#define W2   14
#define C2   32
#define O2   64
#define KK   9
#define KDIM2 (C2 * KK)   // 288
#define FCK  3136
#define FCN  128

__device__ __forceinline__ float sigmoid2(float v) {
  return 2.0f / (1.0f + expf(-v));
}

struct Bilin {
  int i00, i01, i10, i11;
  float w00, w01, w10, w11;
};

// Matches reference: per-corner validity mask, clipped gather index, weight.
__device__ __forceinline__ Bilin bilin_setup(float py, float px, int H, int W,
                                             float scale) {
  float y0f = floorf(py), x0f = floorf(px);
  int y0 = (int)y0f, x0i = (int)x0f;
  int y1 = y0 + 1, x1 = x0i + 1;
  float wy1 = py - y0f, wx1 = px - x0f;
  float wy0 = 1.0f - wy1, wx0 = 1.0f - wx1;
  bool vy0 = (y0 >= 0) && (y0 < H);
  bool vy1 = (y1 >= 0) && (y1 < H);
  bool vx0 = (x0i >= 0) && (x0i < W);
  bool vx1 = (x1 >= 0) && (x1 < W);
  int cy0 = min(max(y0, 0), H - 1), cy1 = min(max(y1, 0), H - 1);
  int cx0 = min(max(x0i, 0), W - 1), cx1 = min(max(x1, 0), W - 1);
  Bilin r;
  r.i00 = cy0 * W + cx0; r.i01 = cy0 * W + cx1;
  r.i10 = cy1 * W + cx0; r.i11 = cy1 * W + cx1;
  r.w00 = (vy0 && vx0) ? wy0 * wx0 * scale : 0.0f;
  r.w01 = (vy0 && vx1) ? wy0 * wx1 * scale : 0.0f;
  r.w10 = (vy1 && vx0) ? wy1 * wx0 * scale : 0.0f;
  r.w11 = (vy1 && vx1) ? wy1 * wx1 * scale : 0.0f;
  return r;
}

// ---------------------------------------------------------------------------
// K1: block-1 fully fused (C_in = 1): offsets conv + mask conv + bilinear
// gather + 32-wide regular conv. One thread per output pixel.
// ---------------------------------------------------------------------------
__global__ __launch_bounds__(256) void k1_block1(
    const float* __restrict__ x,
    const float* __restrict__ off_w1, const float* __restrict__ off_b1,
    const float* __restrict__ mod_w1, const float* __restrict__ mod_b1,
    const float* __restrict__ reg_w1, float* __restrict__ h1) {
  int idx = blockIdx.x * 256 + threadIdx.x;
  if (idx >= NB * H1 * W1) return;
  int b = idx / (H1 * W1);
  int yx = idx % (H1 * W1);
  int y = yx / W1, xx = yx % W1;
  const float* xb = x + (size_t)b * (H1 * W1);

  float patch[9];
#pragma unroll
  for (int t = 0; t < 9; ++t) {
    int py = y + t / 3 - 1, px = xx + t % 3 - 1;
    patch[t] = (py >= 0 && py < H1 && px >= 0 && px < W1) ? xb[py * W1 + px]
                                                          : 0.0f;
  }
  float out[C1O];
#pragma unroll
  for (int o = 0; o < C1O; ++o) out[o] = 0.0f;

#pragma unroll
  for (int k = 0; k < KK; ++k) {
    float dy = off_b1[2 * k], dx = off_b1[2 * k + 1], mk = mod_b1[k];
#pragma unroll
    for (int t = 0; t < 9; ++t) {
      dy += off_w1[(2 * k) * 9 + t] * patch[t];
      dx += off_w1[(2 * k + 1) * 9 + t] * patch[t];
      mk += mod_w1[k * 9 + t] * patch[t];
    }
    mk = sigmoid2(mk);
    float py = (float)y + (float)(k / 3) - 1.0f + dy;
    float px = (float)xx + (float)(k % 3) - 1.0f + dx;
    Bilin bl = bilin_setup(py, px, H1, W1, mk);
    float val = bl.w00 * xb[bl.i00] + bl.w01 * xb[bl.i01] +
                bl.w10 * xb[bl.i10] + bl.w11 * xb[bl.i11];
#pragma unroll
    for (int o = 0; o < C1O; ++o) out[o] += val * reg_w1[o * 9 + k];
  }
#pragma unroll
  for (int o = 0; o < C1O; ++o)
    h1[((size_t)(b * C1O + o)) * (H1 * W1) + yx] = out[o];
}

// ---------------------------------------------------------------------------
// K2: maxpool2(relu(h1)) -> p1 (512,32,14,14). relu(max) == max(relu).
// ---------------------------------------------------------------------------
__global__ __launch_bounds__(256) void k2_pool1(const float* __restrict__ h1,
                                                float* __restrict__ p1) {
  int idx = blockIdx.x * 256 + threadIdx.x;
  if (idx >= NB * C1O * H2 * W2) return;
  int xw = idx % W2;
  int t = idx / W2;
  int yh = t % H2;
  t /= H2;  // t = b*C1O + c
  const float* src = h1 + (size_t)t * (H1 * W1) + (2 * yh) * W1 + 2 * xw;
  float m = fmaxf(fmaxf(src[0], src[1]), fmaxf(src[W1], src[W1 + 1]));
  p1[idx] = fmaxf(m, 0.0f);
}

// ---------------------------------------------------------------------------
// K3: block-2 offsets (18ch, raw) + mask (9ch, 2*sigmoid) 3x3 conv on p1.
// One thread per (b, oc, y, x); K = 32*9 = 288 MACs each.
// ---------------------------------------------------------------------------
__global__ __launch_bounds__(256) void k3_conv27(
    const float* __restrict__ p1,
    const float* __restrict__ off_w2, const float* __restrict__ off_b2,
    const float* __restrict__ mod_w2, const float* __restrict__ mod_b2,
    float* __restrict__ offmask) {
  int idx = blockIdx.x * 256 + threadIdx.x;
  if (idx >= NB * 27 * H2 * W2) return;
  int b = idx / (27 * H2 * W2);
  int rem = idx % (27 * H2 * W2);
  int oc = rem / (H2 * W2);
  int yx = rem % (H2 * W2);
  int y = yx / W2, xx = yx % W2;
  const float* w;
  float acc;
  if (oc < 18) { w = off_w2 + (size_t)oc * KDIM2; acc = off_b2[oc]; }
  else         { w = mod_w2 + (size_t)(oc - 18) * KDIM2; acc = mod_b2[oc - 18]; }
  const float* xb = p1 + (size_t)b * C2 * (H2 * W2);
  for (int c = 0; c < C2; ++c) {
    const float* xc = xb + c * (H2 * W2);
    const float* wc = w + c * 9;
#pragma unroll
    for (int ky = 0; ky < 3; ++ky) {
      int yy = y + ky - 1;
      if (yy < 0 || yy >= H2) continue;
#pragma unroll
      for (int kx = 0; kx < 3; ++kx) {
        int xc2 = xx + kx - 1;
        if (xc2 < 0 || xc2 >= W2) continue;
        acc += wc[ky * 3 + kx] * xc[yy * W2 + xc2];
      }
    }
  }
  if (oc >= 18) acc = sigmoid2(acc);
  offmask[idx] = acc;
}

// ---------------------------------------------------------------------------
// K4: block-2 main: deformed im2col tile (16 pixels x 288) into LDS, then
// 4 waves x WMMA f32 16x16x4 over K=288 -> 16x64 outputs, ReLU fused.
// GEMM: M=100352, N=64, K=288. EXEC stays all-ones (no divergence).
// ---------------------------------------------------------------------------
__global__ __launch_bounds__(128) void k4_block2(
    const float* __restrict__ p1, const float* __restrict__ offmask,
    const float* __restrict__ reg_w2, float* __restrict__ h2) {
  __shared__ float Atile[16][KDIM2 + 4];  // +4 pad vs LDS bank conflicts
  int tid = threadIdx.x;
  int m0 = blockIdx.x * 16;

  // Build A tile: one task per (pixel-in-tile, tap k); each writes 32 c's.
  for (int task = tid; task < 16 * KK; task += 128) {
    int i = task / KK, k = task % KK;
    int m = m0 + i;
    int b = m / (H2 * W2);
    int yx = m % (H2 * W2);
    int y = yx / W2, xx = yx % W2;
    const float* om = offmask + (size_t)b * 27 * (H2 * W2) + yx;
    float dy = om[(2 * k) * (H2 * W2)];
    float dx = om[(2 * k + 1) * (H2 * W2)];
    float mk = om[(18 + k) * (H2 * W2)];
    float py = (float)y + (float)(k / 3) - 1.0f + dy;
    float px = (float)xx + (float)(k % 3) - 1.0f + dx;
    Bilin bl = bilin_setup(py, px, H2, W2, mk);
    const float* xb = p1 + (size_t)b * C2 * (H2 * W2);
    for (int c = 0; c < C2; ++c) {
      const float* xc = xb + c * (H2 * W2);
      Atile[i][c * KK + k] = bl.w00 * xc[bl.i00] + bl.w01 * xc[bl.i01] +
                             bl.w10 * xc[bl.i10] + bl.w11 * xc[bl.i11];
    }
  }
  __syncthreads();

  int lane = tid & 31;
  int wave = tid >> 5;        // wave -> 16-column N tile
  int n0 = wave * 16;
  int mL = lane & 15;
  int kOff = (lane >> 4) * 2; // A/B frag layout: lane-half selects K pair
  const float* bRow = reg_w2 + (size_t)(n0 + mL) * KDIM2;  // B[n][k] row-major
  v8f acc = {};
  for (int k = 0; k < KDIM2; k += 4) {
    v2f a, bb;
    a.x = Atile[mL][k + kOff];       // A[m][k+kOff]   (VGPR0)
    a.y = Atile[mL][k + kOff + 1];   // A[m][k+kOff+1] (VGPR1)
    bb.x = bRow[k + kOff];           // B[k+kOff][n]
    bb.y = bRow[k + kOff + 1];       // B[k+kOff+1][n]
    acc = __builtin_amdgcn_wmma_f32_16x16x4_f32(
        false, a, false, bb, (short)0, acc, false, false);
  }
  // D layout: VGPR i -> rows i (lanes 0-15) / i+8 (lanes 16-31), col = lane&15
  int nc = n0 + (lane & 15);
  int mBase = (lane < 16) ? 0 : 8;
#pragma unroll
  for (int i = 0; i < 8; ++i) {
    int m = m0 + mBase + i;
    int b = m / (H2 * W2);
    int yx = m % (H2 * W2);
    h2[((size_t)b * O2 + nc) * (H2 * W2) + yx] = fmaxf(acc[i], 0.0f);  // ReLU
  }
}

// ---------------------------------------------------------------------------
// K5: maxpool2 on (already-ReLU'd) h2 -> p2 flattened (512, 3136).
// ---------------------------------------------------------------------------
__global__ __launch_bounds__(256) void k5_pool2(const float* __restrict__ h2,
                                                float* __restrict__ p2) {
  int idx = blockIdx.x * 256 + threadIdx.x;
  if (idx >= NB * O2 * 7 * 7) return;
  int xw = idx % 7;
  int t = idx / 7;
  int yh = t % 7;
  t /= 7;  // t = b*64 + c
  const float* src = h2 + (size_t)t * (H2 * W2) + (2 * yh) * W2 + 2 * xw;
  p2[idx] = fmaxf(fmaxf(src[0], src[1]), fmaxf(src[W2], src[W2 + 1]));
}

// ---------------------------------------------------------------------------
// K6: FC1 via WMMA f32 16x16x4: (512 x 3136) @ (3136 x 128) + bias, ReLU.
// Block = 4 waves, each one 16x16 tile; grid (512/16, 128/64).
// ---------------------------------------------------------------------------
__global__ __launch_bounds__(128) void k6_fc1(
    const float* __restrict__ p2, const float* __restrict__ fc1_w,
    const float* __restrict__ fc1_b, float* __restrict__ fc1o) {
  int tid = threadIdx.x;
  int lane = tid & 31;
  int wave = tid >> 5;
  int m0 = blockIdx.x * 16;
  int n0 = blockIdx.y * 64 + wave * 16;
  int mL = lane & 15;
  int kOff = (lane >> 4) * 2;
  const float* aRow = p2 + (size_t)(m0 + mL) * FCK;
  const float* bRow = fc1_w + (size_t)(n0 + mL) * FCK;  // out = h @ W^T
  v8f acc = {};
  for (int k = 0; k < FCK; k += 4) {
    v2f a, bb;
    a.x = aRow[k + kOff];
    a.y = aRow[k + kOff + 1];
    bb.x = bRow[k + kOff];
    bb.y = bRow[k + kOff + 1];
    acc = __builtin_amdgcn_wmma_f32_16x16x4_f32(
        false, a, false, bb, (short)0, acc, false, false);
  }
  int nc = n0 + (lane & 15);
  int mBase = m0 + ((lane < 16) ? 0 : 8);
  float bias = fc1_b[nc];
#pragma unroll
  for (int i = 0; i < 8; ++i)
    fc1o[(size_t)(mBase + i) * FCN + nc] = fmaxf(acc[i] + bias, 0.0f);
}

// ---------------------------------------------------------------------------
// K7: FC2 (512 x 128 x 10) + bias. Tiny: one thread per output.
// ---------------------------------------------------------------------------
__global__ __launch_bounds__(256) void k7_fc2(
    const float* __restrict__ fc1o, const float* __restrict__ fc2_w,
    const float* __restrict__ fc2_b, float* __restrict__ out) {
  int idx = blockIdx.x * 256 + threadIdx.x;
  if (idx >= NB * 10) return;
  int n = idx % 10, m = idx / 10;
  const float* a = fc1o + (size_t)m * FCN;
  const float* w = fc2_w + (size_t)n * FCN;
  float acc = fc2_b[n];
  for (int k = 0; k < FCN; ++k) acc += a[k] * w[k];
  out[idx] = acc;
}

// ---------------------------------------------------------------------------
extern "C" void kernel_launch(void* const* d_in, const int* in_sizes, int n_in,
                              void* d_out, int out_size, void* d_ws,
                              size_t ws_size, hipStream_t stream) {
  const float* x      = (const float*)d_in[0];
  const float* off_w1 = (const float*)d_in[1];
  const float* off_b1 = (const float*)d_in[2];
  const float* mod_w1 = (const float*)d_in[3];
  const float* mod_b1 = (const float*)d_in[4];
  const float* reg_w1 = (const float*)d_in[5];
  const float* off_w2 = (const float*)d_in[6];
  const float* off_b2 = (const float*)d_in[7];
  const float* mod_w2 = (const float*)d_in[8];
  const float* mod_b2 = (const float*)d_in[9];
  const float* reg_w2 = (const float*)d_in[10];
  const float* fc1_w  = (const float*)d_in[11];
  const float* fc1_b  = (const float*)d_in[12];
  const float* fc2_w  = (const float*)d_in[13];
  const float* fc2_b  = (const float*)d_in[14];
  float* out = (float*)d_out;

  // Workspace layout (floats). h2 reuses the h1 region (dead after K2).
  float* ws   = (float*)d_ws;
  float* h1   = ws;                        // 512*32*784  = 12,845,056
  float* h2   = ws;                        // 512*64*196  =  6,422,528 (reuse)
  float* p1   = ws + 12845056;             // 512*32*196  =  3,211,264
  float* om2  = p1 + 3211264;              // 512*27*196  =  2,709,504
  float* p2   = om2 + 2709504;             // 512*3136    =  1,605,632
  float* fc1o = p2 + 1605632;              // 512*128     =     65,536

  k1_block1<<<(NB * H1 * W1 + 255) / 256, 256, 0, stream>>>(
      x, off_w1, off_b1, mod_w1, mod_b1, reg_w1, h1);
  k2_pool1<<<(NB * C1O * H2 * W2 + 255) / 256, 256, 0, stream>>>(h1, p1);
  k3_conv27<<<(NB * 27 * H2 * W2 + 255) / 256, 256, 0, stream>>>(
      p1, off_w2, off_b2, mod_w2, mod_b2, om2);
  k4_block2<<<(NB * H2 * W2) / 16, 128, 0, stream>>>(p1, om2, reg_w2, h2);
  k5_pool2<<<(NB * O2 * 7 * 7 + 255) / 256, 256, 0, stream>>>(h2, p2);
  k6_fc1<<<dim3(NB / 16, FCN / 64), 128, 0, stream>>>(p2, fc1_w, fc1_b, fc1o);
  k7_fc2<<<(NB * 10 + 255) / 256, 256, 0, stream>>>(fc1o, fc2_w, fc2_b, out);
}